// ModelNew_80908593922571
// MI455X (gfx1250) — compile-verified
//
#include <hip/hip_runtime.h>

typedef __attribute__((ext_vector_type(2))) float v2f;
typedef __attribute__((ext_vector_type(8))) float v8f;

#define CI      32
#define CO      64
#define IND     16
#define INH     32
#define INW     32
#define TILE_D  5
#define TILE_H  5
#define TILE_W  34              // iw in [-1,32] zero-padded
#define SPAT    (TILE_D * TILE_H * TILE_W)   // 850 spatial slots per ci
// ci pairs interleaved innermost; pair-plane padded so that the +1 pair step
// used by lanes 16..31 lands on the opposite 32 LDS banks (1760 % 64 == 32).
#define PPLANE  1760                          // words per ci-pair (2*850=1700 data + 60 pad)
#define NPAIR   (CI / 2)                      // 16
#define XTILE_W (NPAIR * PPLANE)              // 28160 words
#define CMAXN   (10 * CO)                     // 640 words
#define SMEM_BYTES ((XTILE_W + CMAXN) * sizeof(float))

__device__ __forceinline__ void atomic_max_f32_lds(float* p, float v) {
    // monotone float ordering: positives via signed max, negatives via unsigned min
    if (v >= 0.0f) atomicMax((int*)p, __float_as_int(v));
    else           atomicMin((unsigned int*)p, __float_as_uint(v));
}

__global__ __launch_bounds__(256) void convt3d_pool_sum_kernel(
    const float* __restrict__ x, const float* __restrict__ w,
    const float* __restrict__ bias, float* __restrict__ out)
{
    extern __shared__ float smem[];
    float* xs   = smem;              // [ci/2][id_l][ih_l][iw+1][2] (+pad)
    float* cmax = smem + XTILE_W;    // [ow3][co]

    const int tid = threadIdx.x;
    const int bx  = blockIdx.x;      // 800 blocks = 16 * 5 * 10
    const int n   = bx / 50;
    const int rem = bx % 50;
    const int od3 = rem / 10;
    const int oh3 = rem % 10;

    // ---- stage zero-padded x tile into LDS (ci-pair interleaved) ----
    const int id0 = 3 * od3 - 1;
    const int ih0 = 3 * oh3 - 1;
    for (int idx = tid; idx < CI * SPAT; idx += 256) {
        int iwp = idx % TILE_W;
        int t   = idx / TILE_W;
        int ihl = t % TILE_H; t /= TILE_H;
        int idl = t % TILE_D;
        int ci  = t / TILE_D;
        int id = id0 + idl, ih = ih0 + ihl, iw = iwp - 1;
        float v = 0.0f;
        if ((unsigned)id < (unsigned)IND && (unsigned)ih < (unsigned)INH &&
            (unsigned)iw < (unsigned)INW)
            v = x[(((n * CI + ci) * IND + id) * INH + ih) * INW + iw];
        const int spat = (idl * TILE_H + ihl) * TILE_W + iwp;
        xs[(ci >> 1) * PPLANE + spat * 2 + (ci & 1)] = v;
    }
    for (int i = tid; i < CMAXN; i += 256)
        cmax[i] = -__builtin_inff();
    __syncthreads();

    // ---- wave tiling: 8 waves = 2 M-tiles (ow) x 4 N-tiles (co) ----
    const int lane  = tid & 31;            // wave32
    const int wv    = tid >> 5;
    const int mtile = wv & 1;
    const int ntile = wv >> 1;
    const int mrow  = lane & 15;
    const int kp    = lane >> 4;           // selects K pair {2kp, 2kp+1}
    const int jbase = 16 * mtile;          // j index of ow: ow = pw + 2*j
    const int co    = 16 * ntile + mrow;   // this lane's output channel (B/C column)

    const v8f vzero = {0.f, 0.f, 0.f, 0.f, 0.f, 0.f, 0.f, 0.f};

    for (int pc = 0; pc < 8; ++pc) {       // 8 stride-2 parity classes
        const int pd = (pc >> 2) & 1, ph = (pc >> 1) & 1, pw = pc & 1;
        const int ntd = 3 - pd, nth = 3 - ph, ntw = 3 - pw;

        v8f acc[9];
        #pragma unroll
        for (int p = 0; p < 9; ++p) acc[p] = vzero;

        for (int td = 0; td < ntd; ++td) {
            const int kd = pd + 2 * td;
            for (int th = 0; th < nth; ++th) {
                const int kh = ph + 2 * th;
                for (int tw = 0; tw < ntw; ++tw) {
                    const int kw  = pw + 2 * tw;
                    const int tap = (kd * 5 + kh) * 5 + kw;
                    const int iwp = jbase + mrow + 2 - tw;   // padded LDS iw index
                    for (int ci0 = 0; ci0 < CI; ci0 += 4) {
                        const int cib  = ci0 + 2 * kp;       // even
                        const int pair = (ci0 >> 1) + kp;
                        const int widx = (cib * CO + co) * 125 + tap;
                        v2f b;
                        b.x = w[widx];
                        b.y = w[widx + CO * 125];
                        #pragma unroll
                        for (int p = 0; p < 9; ++p) {
                            const int odp = p / 3, ohp = p % 3;
                            const int idl = odp + 2 - td;    // always in [0,4]
                            const int ihl = ohp + 2 - th;
                            const int spat =
                                (idl * TILE_H + ihl) * TILE_W + iwp;
                            // one conflict-free ds_load_b64: {ci=cib, ci=cib+1}
                            v2f a = *(const v2f*)(xs + pair * PPLANE + spat * 2);
                            acc[p] = __builtin_amdgcn_wmma_f32_16x16x4_f32(
                                false, a, false, b, (short)0, acc[p],
                                false, false);
                        }
                    }
                }
            }
        }

        // ---- fold this class's conv values into the 6x6x6 max (per ow3, co) ----
        #pragma unroll
        for (int p = 0; p < 9; ++p) {
            #pragma unroll
            for (int v = 0; v < 8; ++v) {
                const int jc = jbase + v + 8 * kp;   // C layout: M = v + 8*(lane>=16)
                if (jc < 30) {                        // ow3 = jc/3; jc 30,31 unused
                    atomic_max_f32_lds(&cmax[(jc / 3) * CO + co], acc[p][v]);
                }
            }
        }
    }
    __syncthreads();

    // ---- channel sum (+bias shift commutes with max) ----
    if (tid < 10) {
        float s = 0.0f;
        for (int c = 0; c < CO; ++c)
            s += cmax[tid * CO + c] + bias[c];
        out[((n * 5 + od3) * 10 + oh3) * 10 + tid] = s;
    }
}

extern "C" void kernel_launch(void* const* d_in, const int* in_sizes, int n_in,
                              void* d_out, int out_size, void* d_ws, size_t ws_size,
                              hipStream_t stream) {
    const float* x    = (const float*)d_in[0];
    const float* wgt  = (const float*)d_in[1];
    const float* bias = (const float*)d_in[2];
    float* out        = (float*)d_out;
    (void)in_sizes; (void)n_in; (void)out_size; (void)d_ws; (void)ws_size;

    convt3d_pool_sum_kernel<<<dim3(16 * 5 * 10), dim3(256), SMEM_BYTES, stream>>>(
        x, wgt, bias, out);
}